// SAGGNN_76184129896625
// MI455X (gfx1250) — compile-verified
//
#include <hip/hip_runtime.h>

#define N_NODES 8192
#define N_EDGES 262144
#define DIM     128

typedef __attribute__((ext_vector_type(2))) float v2f;
typedef __attribute__((ext_vector_type(4))) float v4f;
typedef __attribute__((ext_vector_type(8))) float v8f;

// ---------------------------------------------------------------------------
// Fp32 WMMA GEMM:  C[M x 128] = A[M x 128] * B[128 x 128] (+ bias)
// One 16x16 output tile per wave, K stepped by 4 via V_WMMA_F32_16X16X4_F32.
// A-frag (16x4 f32): lane l holds row m=l&15, K = (l>>4)*2 .. +1  -> 2 VGPRs
// B-frag (4x16 f32): lane l holds col n=l&15, K = (l>>4)*2 .. +1  -> 2 VGPRs
// C/D (16x16 f32):   vgpr r, lane l -> (m = r + (l>>4)*8, n = l&15)
// Full B (64KB) is staged in LDS and shared by all 8 waves of the block.
// ---------------------------------------------------------------------------
__global__ __launch_bounds__(256) void gemm128_wmma_f32(
    const float* __restrict__ A, const float* __restrict__ B,
    const float* __restrict__ bias, float* __restrict__ C, int M)
{
    __shared__ float sB[DIM * DIM];
    // cooperative 64KB load of B into LDS (float4 per thread-iter)
    for (int i = threadIdx.x; i < DIM * DIM / 4; i += 256) {
        ((v4f*)sB)[i] = ((const v4f*)B)[i];
    }
    __syncthreads();

    const int wave = threadIdx.x >> 5;
    const int lane = threadIdx.x & 31;
    const int gw   = blockIdx.x * 8 + wave;      // global wave id
    const int tiles_n = DIM / 16;                // 8
    const int tm = gw / tiles_n;
    const int tn = gw % tiles_n;
    if (tm * 16 >= M) return;                    // wave-uniform guard

    const int mn   = lane & 15;                  // row of A-frag == col of B-frag
    const int koff = (lane >> 4) << 1;           // 0 or 2

    const float* Arow = A + (size_t)(tm * 16 + mn) * DIM;
    const float* Bcol = sB + tn * 16 + mn;

    v8f acc = {};
    #pragma unroll 4
    for (int k = 0; k < DIM; k += 4) {
        v2f av, bv;
        av.x = Arow[k + koff];
        av.y = Arow[k + koff + 1];
        bv.x = Bcol[(k + koff) * DIM];
        bv.y = Bcol[(k + koff + 1) * DIM];
        acc = __builtin_amdgcn_wmma_f32_16x16x4_f32(
            /*neg_a=*/false, av, /*neg_b=*/false, bv,
            /*c_mod=*/(short)0, acc, /*reuse_a=*/false, /*reuse_b=*/false);
    }

    const float bcol = bias ? bias[tn * 16 + mn] : 0.0f;
    const int mbase  = tm * 16 + ((lane >> 4) << 3);
    #pragma unroll
    for (int r = 0; r < 8; ++r) {
        C[(size_t)(mbase + r) * DIM + tn * 16 + mn] = acc[r] + bcol;
    }
}

// ---------------------------------------------------------------------------
// Degree / normalization
// ---------------------------------------------------------------------------
__global__ void deg_init(float* __restrict__ deg, int n) {
    int i = blockIdx.x * blockDim.x + threadIdx.x;
    if (i < n) deg[i] = 1.0f;                    // self loop
}

__global__ void deg_count(const int* __restrict__ dst, float* __restrict__ deg, int ne) {
    int e = blockIdx.x * blockDim.x + threadIdx.x;
    if (e < ne) atomicAdd(&deg[dst[e]], 1.0f);
}

__global__ void make_dinv(const float* __restrict__ deg, float* __restrict__ dinv, int n) {
    int i = blockIdx.x * blockDim.x + threadIdx.x;
    if (i < n) dinv[i] = rsqrtf(deg[i]);
}

// ---------------------------------------------------------------------------
// Aggregation: out[v] = h[v]/deg[v] + sum_{e:dst=v} h[src[e]] * dinv[s]*dinv[v]
// agg_self initializes out fully (self-loop term), edge_scatter atomically adds.
// 32 lanes per node/edge, float4 per lane covers the 128-wide feature row.
// ---------------------------------------------------------------------------
__global__ void agg_self(const float* __restrict__ h, const float* __restrict__ dinv,
                         float* __restrict__ out, int n) {
    int idx = blockIdx.x * blockDim.x + threadIdx.x;
    int i = idx >> 5, f4 = idx & 31;
    if (i >= n) return;
    float s = dinv[i] * dinv[i];                 // == 1/deg[i]
    v4f v = ((const v4f*)(h + (size_t)i * DIM))[f4];
    v *= s;
    ((v4f*)(out + (size_t)i * DIM))[f4] = v;
}

__global__ void edge_scatter(const int* __restrict__ src, const int* __restrict__ dst,
                             const float* __restrict__ h, const float* __restrict__ dinv,
                             float* __restrict__ out, int ne) {
    int idx = blockIdx.x * blockDim.x + threadIdx.x;
    int e = idx >> 5, f4 = idx & 31;
    if (e >= ne) return;
    int s = src[e], d = dst[e];
    float nrm = dinv[s] * dinv[d];
    v4f v = ((const v4f*)(h + (size_t)s * DIM))[f4];
    float* o = out + (size_t)d * DIM + f4 * 4;
    atomicAdd(o + 0, v.x * nrm);
    atomicAdd(o + 1, v.y * nrm);
    atomicAdd(o + 2, v.z * nrm);
    atomicAdd(o + 3, v.w * nrm);
}

__global__ void bias_relu(float* __restrict__ h, const float* __restrict__ b, int total) {
    int idx = blockIdx.x * blockDim.x + threadIdx.x;
    if (idx < total) h[idx] = fmaxf(h[idx] + b[idx & (DIM - 1)], 0.0f);
}

// ---------------------------------------------------------------------------
// Output head: a[i] = h[i,:].w_i  ,  c[i] = h[i,:].w_j
// ---------------------------------------------------------------------------
__global__ void node_dots(const float* __restrict__ h, const float* __restrict__ wedge,
                          float* __restrict__ a, float* __restrict__ c, int n) {
    int i = blockIdx.x * blockDim.x + threadIdx.x;
    if (i >= n) return;
    const float* r = h + (size_t)i * DIM;
    float sa = 0.0f, sc = 0.0f;
    #pragma unroll 4
    for (int k = 0; k < DIM; ++k) {
        float v = r[k];
        sa = fmaf(v, wedge[k], sa);
        sc = fmaf(v, wedge[DIM + k], sc);
    }
    a[i] = sa;
    c[i] = sc;
}

// edge_logits[i][j] = a[i] + c[j] + b_edge
// 256MB write-once stream > 192MB L2  ->  non-temporal b128 stores.
__global__ void outer_sum(const float* __restrict__ a, const float* __restrict__ c,
                          const float* __restrict__ b_edge, float* __restrict__ out, int n) {
    int idx = blockIdx.x * blockDim.x + threadIdx.x;   // n * n/4 threads
    int nq = n >> 2;
    int i = idx / nq, j4 = idx % nq;
    if (i >= n) return;
    float ai = a[i] + b_edge[0];
    v4f cv = ((const v4f*)c)[j4];                      // 32KB, stays hot in cache (RT)
    v4f o  = cv + ai;
    __builtin_nontemporal_store(o, ((v4f*)out) + (size_t)i * nq + j4);
}

// ---------------------------------------------------------------------------
extern "C" void kernel_launch(void* const* d_in, const int* in_sizes, int n_in,
                              void* d_out, int out_size, void* d_ws, size_t ws_size,
                              hipStream_t stream) {
    const float* x      = (const float*)d_in[0];
    const int*   ei     = (const int*)  d_in[1];
    const float* W1     = (const float*)d_in[2];
    const float* b1     = (const float*)d_in[3];
    const float* W2     = (const float*)d_in[4];
    const float* b2     = (const float*)d_in[5];
    const float* W_out  = (const float*)d_in[6];
    const float* b_out  = (const float*)d_in[7];
    const float* W_edge = (const float*)d_in[8];
    const float* b_edge = (const float*)d_in[9];
    float* out = (float*)d_out;

    const int* src = ei;
    const int* dst = ei + N_EDGES;

    // workspace layout (floats): deg | dinv | a | c | h | agg
    float* ws   = (float*)d_ws;
    float* deg  = ws;
    float* dinv = ws + N_NODES;
    float* av   = ws + 2 * N_NODES;
    float* cv   = ws + 3 * N_NODES;
    float* h    = ws + 4 * N_NODES;                     // N_NODES*DIM
    float* agg  = h + (size_t)N_NODES * DIM;            // N_NODES*DIM

    const int gemm_blocks = N_NODES / 16;               // 512 (8 waves/block, 8 N-tiles/row)

    // degree + normalization (independent of GEMM; issue first)
    deg_init <<<(N_NODES + 255) / 256, 256, 0, stream>>>(deg, N_NODES);
    deg_count<<<(N_EDGES + 255) / 256, 256, 0, stream>>>(dst, deg, N_EDGES);
    make_dinv<<<(N_NODES + 255) / 256, 256, 0, stream>>>(deg, dinv, N_NODES);

    // layer 1: h = x @ W1 ; agg = norm-aggregate(h) ; relu(agg + b1)
    gemm128_wmma_f32<<<gemm_blocks, 256, 0, stream>>>(x, W1, nullptr, h, N_NODES);
    agg_self    <<<(N_NODES * 32) / 256, 256, 0, stream>>>(h, dinv, agg, N_NODES);
    edge_scatter<<<(N_EDGES * 32) / 256, 256, 0, stream>>>(src, dst, h, dinv, agg, N_EDGES);
    bias_relu   <<<(N_NODES * DIM) / 256, 256, 0, stream>>>(agg, b1, N_NODES * DIM);

    // layer 2: h = agg @ W2 ; agg = norm-aggregate(h) ; relu(agg + b2)
    gemm128_wmma_f32<<<gemm_blocks, 256, 0, stream>>>(agg, W2, nullptr, h, N_NODES);
    agg_self    <<<(N_NODES * 32) / 256, 256, 0, stream>>>(h, dinv, agg, N_NODES);
    edge_scatter<<<(N_EDGES * 32) / 256, 256, 0, stream>>>(src, dst, h, dinv, agg, N_EDGES);
    bias_relu   <<<(N_NODES * DIM) / 256, 256, 0, stream>>>(agg, b2, N_NODES * DIM);

    // output projection: h = agg @ W_out + b_out
    gemm128_wmma_f32<<<gemm_blocks, 256, 0, stream>>>(agg, W_out, b_out, h, N_NODES);

    // a/c dot products, then 8192x8192 outer sum into d_out
    node_dots<<<(N_NODES + 255) / 256, 256, 0, stream>>>(h, W_edge, av, cv, N_NODES);
    outer_sum<<<(N_NODES * (N_NODES / 4)) / 256, 256, 0, stream>>>(av, cv, b_edge, out, N_NODES);
}